// BiCutLoss_72267119723373
// MI455X (gfx1250) — compile-verified
//
#include <hip/hip_runtime.h>

// Reference: sum over [B=8192, L=2048] pairs of
//   label==1 ? 0.7f*out[...,0] : 1.3f*out[...,1]
// divided by 8192. Pure bandwidth-bound streaming reduction (~268 MB @ 23.3 TB/s).
//
// Per-wave accumulation is done with V_WMMA_F32_16X16X4_F32 against an
// all-ones B matrix: D = A*1 + C accumulates 64 fp32 values per WMMA at full
// precision. All 16 columns of D hold identical sums, so the result is 16x the
// true sum; the 1/16 is folded into the final scale (exact power of two).

typedef __attribute__((ext_vector_type(2))) float v2f;
typedef __attribute__((ext_vector_type(8))) float v8f;

#define NBLOCKS 2048
#define NTHREADS 256

__global__ __launch_bounds__(NTHREADS)
void bicut_partial_kernel(const float4* __restrict__ out4,
                          const int4* __restrict__ lab4,
                          float* __restrict__ partial,
                          int niter) {
    const int tid = blockIdx.x * NTHREADS + threadIdx.x;
    const int total_threads = NBLOCKS * NTHREADS;

    const v2f ones = {1.0f, 1.0f};
    v8f acc = {};  // 16x16 f32 accumulator (8 VGPRs)

    for (int it = 0; it < niter; ++it) {
        // 4 pairs per thread per trip, fully coalesced 16B loads.
        const int p = it * (total_threads * 4) + tid * 4;  // pair index, mult of 4
        const int q = p >> 1;                              // float4 / int4 index

        const float4 oa = out4[q];       // pairs p, p+1
        const float4 ob = out4[q + 1];   // pairs p+2, p+3
        const int4  la = lab4[q];        // int64 labels p, p+1 (lo,hi,lo,hi)
        const int4  lb = lab4[q + 1];    // int64 labels p+2, p+3

        const float w0 = (la.x == 1 && la.y == 0) ? 0.7f * oa.x : 1.3f * oa.y;
        const float w1 = (la.z == 1 && la.w == 0) ? 0.7f * oa.z : 1.3f * oa.w;
        const float w2 = (lb.x == 1 && lb.y == 0) ? 0.7f * ob.x : 1.3f * ob.y;
        const float w3 = (lb.z == 1 && lb.w == 0) ? 0.7f * ob.z : 1.3f * ob.w;

        // Two 16x4 fp32 A tiles per wave per trip (2 VGPRs/lane each).
        const v2f a0 = {w0, w1};
        const v2f a1 = {w2, w3};
        acc = __builtin_amdgcn_wmma_f32_16x16x4_f32(
            false, a0, false, ones, (short)0, acc, false, false);
        acc = __builtin_amdgcn_wmma_f32_16x16x4_f32(
            false, a1, false, ones, (short)0, acc, false, false);
    }

    // Per-lane partial: sum this lane's 8 D-matrix VGPRs.
    float s = ((acc[0] + acc[1]) + (acc[2] + acc[3])) +
              ((acc[4] + acc[5]) + (acc[6] + acc[7]));

    // Deterministic block tree reduction in LDS.
    __shared__ float red[NTHREADS];
    red[threadIdx.x] = s;
    __syncthreads();
    for (int off = NTHREADS / 2; off > 0; off >>= 1) {
        if (threadIdx.x < off) red[threadIdx.x] += red[threadIdx.x + off];
        __syncthreads();
    }
    if (threadIdx.x == 0) partial[blockIdx.x] = red[0];
}

__global__ __launch_bounds__(NTHREADS)
void bicut_final_kernel(const float* __restrict__ partial,
                        float* __restrict__ out,
                        float scale) {
    __shared__ float red[NTHREADS];
    float s = 0.0f;
    for (int i = threadIdx.x; i < NBLOCKS; i += NTHREADS) s += partial[i];
    red[threadIdx.x] = s;
    __syncthreads();
    for (int off = NTHREADS / 2; off > 0; off >>= 1) {
        if (threadIdx.x < off) red[threadIdx.x] += red[threadIdx.x + off];
        __syncthreads();
    }
    if (threadIdx.x == 0) out[0] = red[0] * scale;
}

extern "C" void kernel_launch(void* const* d_in, const int* in_sizes, int n_in,
                              void* d_out, int out_size, void* d_ws, size_t ws_size,
                              hipStream_t stream) {
    const float4* out4 = (const float4*)d_in[0];  // [B, L, 2] fp32
    const int4*   lab4 = (const int4*)d_in[1];    // [B, L] int64
    float* out = (float*)d_out;
    float* partial = (float*)d_ws;                // NBLOCKS floats

    const long long n_pairs = (long long)in_sizes[1];             // B*L = 2^24
    const long long per_sweep = (long long)NBLOCKS * NTHREADS * 4; // pairs/trip
    const int niter = (int)(n_pairs / per_sweep);                 // divides evenly

    // WMMA-with-ones replicates each value into 16 columns -> 16x oversum.
    const float scale = 1.0f / (16.0f * 8192.0f);

    bicut_partial_kernel<<<NBLOCKS, NTHREADS, 0, stream>>>(out4, lab4, partial, niter);
    bicut_final_kernel<<<1, NTHREADS, 0, stream>>>(partial, out, scale);
}